// Voxelization_44074954391645
// MI455X (gfx1250) — compile-verified
//
#include <hip/hip_runtime.h>

// Voxelization scatter-mean for MI455X (gfx1250, wave32).
// features [16,64,65536] f32, coords [16,65536,3] f32, search_area [16,3] f32
// -> out [16,64,20,20,20] f32.
//
// CDNA5 strategy: per-workgroup private voxel accumulator in 160KB dynamic
// LDS (two workgroups co-resident per 320KB WGP) using ds_add_f32 hardware
// LDS float atomics; 128-bit coalesced loads along N; b128 vectorized
// write-out dividing by max(count,1). HBM traffic = read-once 268MB feature
// stream (~13.5us roofline at 23.3 TB/s); idx/coords are L2-resident.

#define GRIDSZ 20
#define RVOX   8000        // 20^3
#define BB     16
#define CC     64
#define NN     65536
#define CPG    4           // channels per workgroup (LDS = 5*8000*4 = 160000B)
#define NGRP   (CC / CPG)  // 16 channel groups -> 256 workgroups total
#define ACC_THREADS 1024
#define LDS_FLOATS ((CPG + 1) * RVOX)           // 4*8000 sums + 8000 counts
#define LDS_BYTES  (LDS_FLOATS * sizeof(float)) // 160000 B; 2 WGs fit in 320KB WGP

__device__ __forceinline__ int voxel_of(const float* __restrict__ c3,
                                        float vs0, float vs1, float vs2) {
  // match reference: floor(coord / voxel_size) + 10, clip to [0,19]
  int ix = (int)(floorf(c3[0] / vs0) + 10.0f);
  int iy = (int)(floorf(c3[1] / vs1) + 10.0f);
  int iz = (int)(floorf(c3[2] / vs2) + 10.0f);
  ix = min(max(ix, 0), GRIDSZ - 1);
  iy = min(max(iy, 0), GRIDSZ - 1);
  iz = min(max(iz, 0), GRIDSZ - 1);
  return ix * (GRIDSZ * GRIDSZ) + iy * GRIDSZ + iz;
}

__global__ __launch_bounds__(256)
void voxel_idx_kernel(const float* __restrict__ coords,
                      const float* __restrict__ search_area,
                      int* __restrict__ idx) {
  int t = blockIdx.x * blockDim.x + threadIdx.x;   // 0 .. B*N-1
  if (t >= BB * NN) return;
  int b = t / NN;
  float v0 = search_area[b * 3 + 0] / (float)GRIDSZ;
  float v1 = search_area[b * 3 + 1] / (float)GRIDSZ;
  float v2 = search_area[b * 3 + 2] / (float)GRIDSZ;
  idx[t] = voxel_of(coords + (size_t)t * 3, v0, v1, v2);
}

template <bool USE_IDX>
__global__ __launch_bounds__(ACC_THREADS)
void voxel_accum_kernel(const float* __restrict__ feat,
                        const int*   __restrict__ idx,
                        const float* __restrict__ coords,
                        const float* __restrict__ search_area,
                        float* __restrict__ out) {
  extern __shared__ float lds[];                 // [CPG][RVOX] sums, [RVOX] counts
  float* sums = lds;
  float* cnts = lds + CPG * RVOX;

  const int b   = blockIdx.x;                    // batch element
  const int c0  = blockIdx.y * CPG;              // first channel of this group
  const int tid = threadIdx.x;

  // Zero the accumulator (b128 LDS stores).
  for (int i = tid * 4; i < LDS_FLOATS; i += ACC_THREADS * 4)
    *(float4*)(lds + i) = make_float4(0.f, 0.f, 0.f, 0.f);
  __syncthreads();

  const float* bfeat = feat + ((size_t)b * CC + c0) * NN;

  float v0 = 0.f, v1 = 0.f, v2 = 0.f;
  const float* bcoords = nullptr;
  const int* bidx = nullptr;
  if (USE_IDX) {
    bidx = idx + (size_t)b * NN;
  } else {
    v0 = search_area[b * 3 + 0] / (float)GRIDSZ;
    v1 = search_area[b * 3 + 1] / (float)GRIDSZ;
    v2 = search_area[b * 3 + 2] / (float)GRIDSZ;
    bcoords = coords + (size_t)b * NN * 3;
  }

  // Accumulate: 4 points per thread per iteration (128-bit coalesced loads,
  // channel strides are compile-time immediates within the 24-bit offset).
  for (int n0 = tid * 4; n0 < NN; n0 += ACC_THREADS * 4) {
    int v[4];
    if (USE_IDX) {
      const int4 v4 = *(const int4*)(bidx + n0);
      v[0] = v4.x; v[1] = v4.y; v[2] = v4.z; v[3] = v4.w;
    } else {
#pragma unroll
      for (int k = 0; k < 4; ++k)
        v[k] = voxel_of(bcoords + (size_t)(n0 + k) * 3, v0, v1, v2);
    }
#pragma unroll
    for (int k = 0; k < 4; ++k) atomicAdd(&cnts[v[k]], 1.0f);   // ds_add_f32
#pragma unroll
    for (int c = 0; c < CPG; ++c) {
      const float4 f4 = *(const float4*)(bfeat + (size_t)c * NN + n0);
      float f[4] = {f4.x, f4.y, f4.z, f4.w};
#pragma unroll
      for (int k = 0; k < 4; ++k)
        atomicAdd(&sums[c * RVOX + v[k]], f[k]);                // ds_add_f32
    }
  }
  __syncthreads();

  // Write-out: b128 LDS reads + b128 coalesced global stores. RVOX % 4 == 0
  // so a 4-group never crosses a channel; 8000 % 64 == 0 keeps the LDS
  // access bank-conflict-free across lanes.
  for (int i = tid * 4; i < CPG * RVOX; i += ACC_THREADS * 4) {
    const int c = i / RVOX;
    const int v = i - c * RVOX;
    const float4 s4 = *(const float4*)(sums + i);
    const float4 n4 = *(const float4*)(cnts + v);
    float4 r;
    r.x = s4.x / fmaxf(n4.x, 1.0f);
    r.y = s4.y / fmaxf(n4.y, 1.0f);
    r.z = s4.z / fmaxf(n4.z, 1.0f);
    r.w = s4.w / fmaxf(n4.w, 1.0f);
    *(float4*)(out + ((size_t)b * CC + (c0 + c)) * RVOX + v) = r;
  }
}

extern "C" void kernel_launch(void* const* d_in, const int* in_sizes, int n_in,
                              void* d_out, int out_size, void* d_ws, size_t ws_size,
                              hipStream_t stream) {
  (void)in_sizes; (void)n_in; (void)out_size;
  const float* features    = (const float*)d_in[0];
  const float* coords      = (const float*)d_in[1];
  const float* search_area = (const float*)d_in[2];
  float* out = (float*)d_out;

  const size_t idx_bytes = (size_t)BB * NN * sizeof(int);
  const bool use_ws = (ws_size >= idx_bytes);

  // Opt in to >64KB dynamic LDS (gfx1250 allows up to 320KB per workgroup).
  (void)hipFuncSetAttribute((const void*)voxel_accum_kernel<true>,
                            hipFuncAttributeMaxDynamicSharedMemorySize, LDS_BYTES);
  (void)hipFuncSetAttribute((const void*)voxel_accum_kernel<false>,
                            hipFuncAttributeMaxDynamicSharedMemorySize, LDS_BYTES);

  dim3 agrid(BB, NGRP);
  if (use_ws) {
    int* idx = (int*)d_ws;
    const int tpts = BB * NN;
    voxel_idx_kernel<<<(tpts + 255) / 256, 256, 0, stream>>>(coords, search_area, idx);
    voxel_accum_kernel<true><<<agrid, ACC_THREADS, LDS_BYTES, stream>>>(
        features, idx, coords, search_area, out);
  } else {
    voxel_accum_kernel<false><<<agrid, ACC_THREADS, LDS_BYTES, stream>>>(
        features, nullptr, coords, search_area, out);
  }
}